// SAGEConv_83099027243350
// MI455X (gfx1250) — compile-verified
//
#include <hip/hip_runtime.h>
#include <hip/hip_bf16.h>

#define DIM      512
#define NNODES   50000
#define NEDGES   400000
#define MTILES   (NNODES / 16)            // 3125 exactly
#define SUPER_M  ((MTILES + 1) / 2)       // 1563 supertiles of 32 rows
#define WAVES_PB 4                        // waves per block (128 threads)
#define MBLOCKS  ((SUPER_M + WAVES_PB - 1) / WAVES_PB)   // 391

typedef __attribute__((ext_vector_type(2))) float v2f;
typedef __attribute__((ext_vector_type(8))) float v8f;

__device__ __forceinline__ v8f wmma4(v2f a, v2f b, v8f c) {
  // V_WMMA_F32_16X16X4_F32 : exact fp32 D = A(16x4) * B(4x16) + C
  return __builtin_amdgcn_wmma_f32_16x16x4_f32(
      /*neg_a=*/false, a, /*neg_b=*/false, b,
      /*c_mod=*/(short)0, c, /*reuse_a=*/false, /*reuse_b=*/false);
}

// One full K=512 pass: 2 M-tiles x 4 N-tiles, 6 b64 loads : 8 WMMAs per step.
__device__ __forceinline__ void mac_pass(const float* __restrict__ A0,
                                         const float* __restrict__ A1,
                                         const float* __restrict__ B0,
                                         const float* __restrict__ B1,
                                         const float* __restrict__ B2,
                                         const float* __restrict__ B3,
                                         v8f acc[8])
{
#pragma unroll 4
  for (int k = 0; k < DIM; k += 4) {
    v2f a0 = *(const v2f*)(A0 + k);
    v2f a1 = *(const v2f*)(A1 + k);
    v2f b0 = *(const v2f*)(B0 + k);
    v2f b1 = *(const v2f*)(B1 + k);
    v2f b2 = *(const v2f*)(B2 + k);
    v2f b3 = *(const v2f*)(B3 + k);
    acc[0] = wmma4(a0, b0, acc[0]);
    acc[1] = wmma4(a0, b1, acc[1]);
    acc[2] = wmma4(a0, b2, acc[2]);
    acc[3] = wmma4(a0, b3, acc[3]);
    acc[4] = wmma4(a1, b0, acc[4]);
    acc[5] = wmma4(a1, b1, acc[5]);
    acc[6] = wmma4(a1, b2, acc[6]);
    acc[7] = wmma4(a1, b3, acc[7]);
  }
}

// Store one 16x64 strip. C/D layout: vgpr r, lane l -> M = r + (l>=16?8:0),
// N = l & 15 (+ 16*j per N-subtile).
__device__ __forceinline__ void store_strip(float* __restrict__ dst,
                                            int mtile, int nbase, int l16, int half,
                                            const v8f* acc,
                                            float b0, float b1, float b2, float b3,
                                            bool relu)
{
  const int mrow = mtile * 16 + 8 * half;
#pragma unroll
  for (int r = 0; r < 8; ++r) {
    float* row = dst + (size_t)(mrow + r) * DIM + nbase + l16;
    float v0 = acc[0][r] + b0;
    float v1 = acc[1][r] + b1;
    float v2 = acc[2][r] + b2;
    float v3 = acc[3][r] + b3;
    if (relu) {
      v0 = fmaxf(v0, 0.0f); v1 = fmaxf(v1, 0.0f);
      v2 = fmaxf(v2, 0.0f); v3 = fmaxf(v3, 0.0f);
    }
    row[ 0] = v0;
    row[16] = v1;
    row[32] = v2;
    row[48] = v3;
  }
}

// ---------------------------------------------------------------------------
// h = relu(feat @ W_pool^T + b_pool)     [NNODES, DIM]
// ---------------------------------------------------------------------------
__global__ __launch_bounds__(128)
void gemm_pool_relu(const float* __restrict__ feat,
                    const float* __restrict__ Wp,   // [DIM, DIM] row-major
                    const float* __restrict__ bp,   // [DIM]
                    float* __restrict__ h)
{
  const int wave = threadIdx.x >> 5;
  const int lane = threadIdx.x & 31;
  const int half = lane >> 4;
  const int l16  = lane & 15;

  const int st = blockIdx.x * WAVES_PB + wave;      // wave-uniform supertile
  if (st >= SUPER_M) return;                        // whole-wave exit, EXEC stays full
  const int m0  = st * 2;
  const int m1  = m0 + 1;
  const bool m1v = (m1 < MTILES);
  const int m1c = m1v ? m1 : m0;                    // clamp: safe garbage compute
  const int nbase = blockIdx.y * 64;

  const float* A0 = feat + (size_t)(m0  * 16 + l16) * DIM + 2 * half;
  const float* A1 = feat + (size_t)(m1c * 16 + l16) * DIM + 2 * half;
  const float* B0 = Wp + (size_t)(nbase +  0 + l16) * DIM + 2 * half;
  const float* B1 = Wp + (size_t)(nbase + 16 + l16) * DIM + 2 * half;
  const float* B2 = Wp + (size_t)(nbase + 32 + l16) * DIM + 2 * half;
  const float* B3 = Wp + (size_t)(nbase + 48 + l16) * DIM + 2 * half;

  v8f acc[8] = {};
  mac_pass(A0, A1, B0, B1, B2, B3, acc);

  const float b0 = bp[nbase +  0 + l16];
  const float b1 = bp[nbase + 16 + l16];
  const float b2 = bp[nbase + 32 + l16];
  const float b3 = bp[nbase + 48 + l16];
  store_strip(h, m0, nbase, l16, half, acc + 0, b0, b1, b2, b3, true);
  if (m1v)
    store_strip(h, m1, nbase, l16, half, acc + 4, b0, b1, b2, b3, true);
}

// ---------------------------------------------------------------------------
// out = feat @ W_self^T + neigh @ W_neigh^T + b_self
// ---------------------------------------------------------------------------
__global__ __launch_bounds__(128)
void gemm_final(const float* __restrict__ feat,
                const float* __restrict__ neigh,
                const float* __restrict__ Ws,
                const float* __restrict__ Wn,
                const float* __restrict__ bs,
                float* __restrict__ out)
{
  const int wave = threadIdx.x >> 5;
  const int lane = threadIdx.x & 31;
  const int half = lane >> 4;
  const int l16  = lane & 15;

  const int st = blockIdx.x * WAVES_PB + wave;
  if (st >= SUPER_M) return;
  const int m0  = st * 2;
  const int m1  = m0 + 1;
  const bool m1v = (m1 < MTILES);
  const int m1c = m1v ? m1 : m0;
  const int nbase = blockIdx.y * 64;

  const size_t a0o = (size_t)(m0  * 16 + l16) * DIM + 2 * half;
  const size_t a1o = (size_t)(m1c * 16 + l16) * DIM + 2 * half;
  const size_t b0o = (size_t)(nbase +  0 + l16) * DIM + 2 * half;
  const size_t b1o = (size_t)(nbase + 16 + l16) * DIM + 2 * half;
  const size_t b2o = (size_t)(nbase + 32 + l16) * DIM + 2 * half;
  const size_t b3o = (size_t)(nbase + 48 + l16) * DIM + 2 * half;

  v8f acc[8] = {};
  // pass 1: feat x W_self^T
  mac_pass(feat + a0o, feat + a1o,
           Ws + b0o, Ws + b1o, Ws + b2o, Ws + b3o, acc);
  // pass 2: neigh x W_neigh^T (accumulate into same tiles)
  mac_pass(neigh + a0o, neigh + a1o,
           Wn + b0o, Wn + b1o, Wn + b2o, Wn + b3o, acc);

  const float b0 = bs[nbase +  0 + l16];
  const float b1 = bs[nbase + 16 + l16];
  const float b2 = bs[nbase + 32 + l16];
  const float b3 = bs[nbase + 48 + l16];
  store_strip(out, m0, nbase, l16, half, acc + 0, b0, b1, b2, b3, false);
  if (m1v)
    store_strip(out, m1, nbase, l16, half, acc + 4, b0, b1, b2, b3, false);
}

// ---------------------------------------------------------------------------
// neigh[dst] = max over incoming edges of h[src].
// h >= 0 (relu) and DGL leaves 0-in-degree rows at 0, so init-to-0 + max is
// exact. For floats >= 0, u32 ordering == float ordering -> u32 atomic max.
// Skip atomics for h==0 (max with 0-initialized dest is a no-op).
// ---------------------------------------------------------------------------
__global__ __launch_bounds__(128)
void scatter_max(const float* __restrict__ h,
                 const int* __restrict__ src,
                 const int* __restrict__ dst,
                 unsigned int* __restrict__ neigh_bits)
{
  const int e = blockIdx.x;
  if (e >= NEDGES) return;
  const int s = src[e];
  const int d = dst[e];
  const int t = threadIdx.x;                 // 128 threads * 4 floats = 512
  const float4 v = ((const float4*)(h + (size_t)s * DIM))[t];
  unsigned int* nb = neigh_bits + (size_t)d * DIM + t * 4;
  if (v.x > 0.0f) atomicMax(nb + 0, __float_as_uint(v.x));
  if (v.y > 0.0f) atomicMax(nb + 1, __float_as_uint(v.y));
  if (v.z > 0.0f) atomicMax(nb + 2, __float_as_uint(v.z));
  if (v.w > 0.0f) atomicMax(nb + 3, __float_as_uint(v.w));
}

__global__ __launch_bounds__(256)
void zero_buf(float4* __restrict__ p, int n4)
{
  const float4 z = {0.0f, 0.0f, 0.0f, 0.0f};
  for (int i = blockIdx.x * blockDim.x + threadIdx.x; i < n4;
       i += gridDim.x * blockDim.x)
    p[i] = z;
}

extern "C" void kernel_launch(void* const* d_in, const int* in_sizes, int n_in,
                              void* d_out, int out_size, void* d_ws, size_t ws_size,
                              hipStream_t stream) {
  const float* feat    = (const float*)d_in[0];
  const float* W_pool  = (const float*)d_in[1];
  const float* b_pool  = (const float*)d_in[2];
  const float* W_neigh = (const float*)d_in[3];
  const float* W_self  = (const float*)d_in[4];
  const float* b_self  = (const float*)d_in[5];
  const int*   src     = (const int*)d_in[6];
  const int*   dst     = (const int*)d_in[7];
  float* out = (float*)d_out;

  float* h     = (float*)d_ws;                       // [NNODES, DIM]
  float* neigh = h + (size_t)NNODES * DIM;           // [NNODES, DIM]

  const int n4 = NNODES * DIM / 4;
  zero_buf<<<2048, 256, 0, stream>>>((float4*)neigh, n4);

  gemm_pool_relu<<<dim3(MBLOCKS, DIM / 64), 128, 0, stream>>>(feat, W_pool, b_pool, h);

  scatter_max<<<NEDGES, 128, 0, stream>>>(h, src, dst, (unsigned int*)neigh);

  gemm_final<<<dim3(MBLOCKS, DIM / 64), 128, 0, stream>>>(feat, neigh, W_self, W_neigh,
                                                          b_self, out);
}